// TransformLoss_11398843203872
// MI455X (gfx1250) — compile-verified
//
#include <hip/hip_runtime.h>

// ---------------------------------------------------------------------------
// TransformLoss: weighted-histogram L1 distance.
//   diff[b] = sum(pre_density where new_mask==b) - sum(origin_density where origin_mask==b)
//   out     = sum_b |diff[b]|
// Memory-bound: 256 MB streamed once -> ~11 us at 23.3 TB/s. No matrix math
// (0 FLOP/byte scatter-reduce, WMMA inapplicable); CDNA5-relevant paths are
// ds_add_f32 (per-wave LDS privatized histograms), native global_atomic_add_f32
// at device scope (guaranteed via inline asm, avoiding any CAS-loop expansion
// that would serialize under 2048-way contention), and NT b128 loads so the
// single-use 256 MB stream does not thrash the 192 MB L2.
// ---------------------------------------------------------------------------

#define NUM_BINS      1025
#define HIST_STRIDE   1056               // padded per-wave copy (multiple of 32)
#define THREADS       256                // 8 wave32s per block
#define WAVES_PER_BLK (THREADS / 32)
#define GRID_BLOCKS   2048

typedef float v4f __attribute__((ext_vector_type(4)));
typedef int   v4i __attribute__((ext_vector_type(4)));

// Native device-scope fp32 far atomic (no return -> STOREcnt, fire-and-forget).
// Default SCOPE=0 would be WGP-coherent only; cross-workgroup accumulation
// requires scope:SCOPE_DEV.
__device__ __forceinline__ void global_fadd_dev(float* p, float v) {
    asm volatile("global_atomic_add_f32 %0, %1, off scope:SCOPE_DEV"
                 :: "v"(p), "v"(v) : "memory");
}

// --- pass 0: zero the global diff histogram (workspace is poisoned 0xAA) ----
__global__ void tl_zero(float* __restrict__ diff) {
    int i = blockIdx.x * blockDim.x + threadIdx.x;
    if (i < NUM_BINS) diff[i] = 0.0f;
}

// --- pass 1: signed diff histogram with per-wave LDS privatization ----------
__global__ __launch_bounds__(THREADS) void tl_hist(
    const v4f* __restrict__ od,   // origin_density
    const v4i* __restrict__ om,   // origin_mask
    const v4f* __restrict__ pd,   // pre_density
    const v4i* __restrict__ nm,   // new_mask
    float* __restrict__ diff,     // [NUM_BINS] global accumulator
    int n4)                       // element count / 4
{
    __shared__ float hist[WAVES_PER_BLK * HIST_STRIDE];

    // zero LDS copies
    for (int i = threadIdx.x; i < WAVES_PER_BLK * HIST_STRIDE; i += THREADS)
        hist[i] = 0.0f;
    __syncthreads();

    // wave32-private histogram slice (gfx1250 is wave32-only)
    float* h = &hist[(threadIdx.x >> 5) * HIST_STRIDE];

    const int stride = gridDim.x * THREADS;
    for (int i = blockIdx.x * THREADS + threadIdx.x; i < n4; i += stride) {
        // single-use stream: non-temporal b128 loads, keep L2 for atomics
        v4f d0 = __builtin_nontemporal_load(od + i);
        v4i m0 = __builtin_nontemporal_load(om + i);
        v4f d1 = __builtin_nontemporal_load(pd + i);
        v4i m1 = __builtin_nontemporal_load(nm + i);

        // masks are guaranteed in [0, 1024] by the reference
        atomicAdd(&h[m0.x], -d0.x);          // ds_add_f32
        atomicAdd(&h[m0.y], -d0.y);
        atomicAdd(&h[m0.z], -d0.z);
        atomicAdd(&h[m0.w], -d0.w);
        atomicAdd(&h[m1.x],  d1.x);
        atomicAdd(&h[m1.y],  d1.y);
        atomicAdd(&h[m1.z],  d1.z);
        atomicAdd(&h[m1.w],  d1.w);
    }
    __syncthreads();

    // flush: sum the 8 wave copies per bin, one global atomic per bin
    for (int b = threadIdx.x; b < NUM_BINS; b += THREADS) {
        float s = 0.0f;
        #pragma unroll
        for (int w = 0; w < WAVES_PER_BLK; ++w)
            s += hist[w * HIST_STRIDE + b];
        if (s != 0.0f)
            global_fadd_dev(&diff[b], s);    // native global_atomic_add_f32
    }
}

// --- pass 2: out[0] = sum_b |diff[b]| ---------------------------------------
__global__ __launch_bounds__(1024) void tl_reduce(const float* __restrict__ diff,
                                                  float* __restrict__ out)
{
    __shared__ float red[32];
    const int tid = threadIdx.x;

    float v = 0.0f;
    for (int i = tid; i < NUM_BINS; i += 1024)
        v += fabsf(diff[i]);

    // wave32 tree reduce
    #pragma unroll
    for (int off = 16; off > 0; off >>= 1)
        v += __shfl_down(v, off, 32);
    if ((tid & 31) == 0) red[tid >> 5] = v;
    __syncthreads();

    if (tid < 32) {
        float w = red[tid];
        #pragma unroll
        for (int off = 16; off > 0; off >>= 1)
            w += __shfl_down(w, off, 32);
        if (tid == 0) out[0] = w;
    }
}

extern "C" void kernel_launch(void* const* d_in, const int* in_sizes, int n_in,
                              void* d_out, int out_size, void* d_ws, size_t ws_size,
                              hipStream_t stream) {
    const v4f* od = (const v4f*)d_in[0];   // origin_density  f32[16.7M]
    const v4i* om = (const v4i*)d_in[1];   // origin_mask     i32[16.7M]
    const v4f* pd = (const v4f*)d_in[2];   // pre_density     f32[16.7M]
    const v4i* nm = (const v4i*)d_in[3];   // new_mask        i32[16.7M]
    float* out  = (float*)d_out;
    float* diff = (float*)d_ws;            // NUM_BINS floats of scratch

    const int n  = in_sizes[0];            // 4096*4096, divisible by 4
    const int n4 = n >> 2;

    tl_zero<<<(NUM_BINS + 255) / 256, 256, 0, stream>>>(diff);
    tl_hist<<<GRID_BLOCKS, THREADS, 0, stream>>>(od, om, pd, nm, diff, n4);
    tl_reduce<<<1, 1024, 0, stream>>>(diff, out);
}